// PatchMemoryBank_18253611008222
// MI455X (gfx1250) — compile-verified
//
#include <hip/hip_runtime.h>

// Farthest-point sampling for N=100000, C=256, S=1000 samples.
// Per step: d2 = ||x||^2 - 2 x.q + ||q||^2 via V_WMMA_F32_16X16X4_F32,
// min-update of per-point min distance, global argmax via last-block pattern.

#define N_PTS   100000
#define C_DIM   256
#define S_SAMP  1000
#define TILES   (N_PTS / 16)        // 6250 exactly
#define TPB     256
#define WPB     (TPB / 32)          // 8 waves per block
#define NBLK    ((TILES + WPB - 1) / WPB)  // 782

typedef __attribute__((ext_vector_type(2))) float v2f;
typedef __attribute__((ext_vector_type(8))) float v8f;

// ---------------------------------------------------------------- init ----
__global__ __launch_bounds__(TPB) void fps_init(const float* __restrict__ feat,
                                                float* __restrict__ normx,
                                                float* __restrict__ mind2,
                                                int* __restrict__ idxI,
                                                int* __restrict__ cnt) {
  int i = blockIdx.x * blockDim.x + threadIdx.x;
  if (i < N_PTS) {
    const float4* row = (const float4*)(feat + (size_t)i * C_DIM);
    float s = 0.0f;
#pragma unroll 8
    for (int k = 0; k < C_DIM / 4; ++k) {
      float4 v = row[k];
      s += v.x * v.x + v.y * v.y + v.z * v.z + v.w * v.w;
    }
    normx[i] = s;
    mind2[i] = 3.0e38f;
  }
  if (i < S_SAMP) cnt[i] = 0;
  if (i == 0) idxI[0] = 0;
}

// ---------------------------------------------------------------- step ----
__global__ __launch_bounds__(TPB) void fps_step(const float* __restrict__ feat,
                                                const float* __restrict__ normx,
                                                float* __restrict__ mind2,
                                                int* __restrict__ idxI,
                                                int* __restrict__ cnt,
                                                float* __restrict__ bval,
                                                int* __restrict__ bidx,
                                                int s) {
  __shared__ __align__(16) float q[C_DIM];
  __shared__ float sv[2 * WPB];
  __shared__ int   si[2 * WPB];
  __shared__ float rv[TPB];
  __shared__ int   ri[TPB];
  __shared__ int   sflag;

  const int tid  = threadIdx.x;
  const int lane = tid & 31;
  const int wave = tid >> 5;
  const int half = lane >> 4;   // 0: lanes 0-15, 1: lanes 16-31
  const int mrow = lane & 15;   // row within 16-point tile for the A fragment

  const int qidx = idxI[s];
  q[tid] = feat[(size_t)qidx * C_DIM + tid];   // TPB == C_DIM
  if (tid < 2 * WPB) { sv[tid] = -1.0f; si[tid] = 0x7fffffff; }
  __syncthreads();

  const float normq = normx[qidx];

  float best  = -1.0f;
  int   besti = 0x7fffffff;

  const int gw     = blockIdx.x * WPB + wave;
  const int stride = gridDim.x * WPB;

  for (int t = gw; t < TILES; t += stride) {
    // A fragment source: lane l -> row (t*16 + l&15), cols k + 2*(l>>4) .. +1
    const float* arow = feat + ((size_t)(t * 16 + mrow)) * C_DIM + half * 2;
    const float* brow = q + half * 2;   // B: query replicated across all 16 cols

    v8f a0 = {}; v8f a1 = {}; v8f a2 = {}; v8f a3 = {};
#pragma unroll
    for (int k = 0; k < C_DIM; k += 16) {
      v2f aA = *(const v2f*)(arow + k);
      v2f bA = *(const v2f*)(brow + k);
      a0 = __builtin_amdgcn_wmma_f32_16x16x4_f32(false, aA, false, bA, (short)0, a0, false, false);
      v2f aB = *(const v2f*)(arow + k + 4);
      v2f bB = *(const v2f*)(brow + k + 4);
      a1 = __builtin_amdgcn_wmma_f32_16x16x4_f32(false, aB, false, bB, (short)0, a1, false, false);
      v2f aC = *(const v2f*)(arow + k + 8);
      v2f bC = *(const v2f*)(brow + k + 8);
      a2 = __builtin_amdgcn_wmma_f32_16x16x4_f32(false, aC, false, bC, (short)0, a2, false, false);
      v2f aD = *(const v2f*)(arow + k + 12);
      v2f bD = *(const v2f*)(brow + k + 12);
      a3 = __builtin_amdgcn_wmma_f32_16x16x4_f32(false, aD, false, bD, (short)0, a3, false, false);
    }
    v8f acc = (a0 + a1) + (a2 + a3);

    // D layout: lane n (0-15) VGPR j holds D[j][n]; lanes 16-31 hold D[8+j][n].
    // B columns are identical, so every lane has the 8 dots for its half.
    if (mrow == 0) {   // lanes 0 and 16 commit rows 0-7 / 8-15 of this tile
      const int base = t * 16 + half * 8;
#pragma unroll
      for (int j = 0; j < 8; ++j) {
        const int p = base + j;
        float d2  = normx[p] - 2.0f * acc[j] + normq;
        float old = mind2[p];
        float nd  = fminf(old, d2);
        mind2[p]  = nd;
        if (nd > best || (nd == best && p < besti)) { best = nd; besti = p; }
      }
    }
  }

  if (mrow == 0) { sv[wave * 2 + half] = best; si[wave * 2 + half] = besti; }
  __syncthreads();

  if (tid == 0) {
    float bv = sv[0]; int bi = si[0];
    for (int k = 1; k < 2 * WPB; ++k)
      if (sv[k] > bv || (sv[k] == bv && si[k] < bi)) { bv = sv[k]; bi = si[k]; }
    bval[blockIdx.x] = bv;
    bidx[blockIdx.x] = bi;
    __threadfence();
    int old = atomicAdd(&cnt[s], 1);
    sflag = (old == (int)gridDim.x - 1) ? 1 : 0;
  }
  __syncthreads();

  if (sflag) {                       // last block: reduce all block results
    __threadfence();
    float bv = -1.0f; int bi = 0x7fffffff;
    for (int k = tid; k < (int)gridDim.x; k += TPB) {
      float v = bval[k]; int ii = bidx[k];
      if (v > bv || (v == bv && ii < bi)) { bv = v; bi = ii; }
    }
    rv[tid] = bv; ri[tid] = bi;
    __syncthreads();
    if (tid == 0) {
      for (int k = 1; k < TPB; ++k)
        if (rv[k] > bv || (rv[k] == bv && ri[k] < bi)) { bv = rv[k]; bi = ri[k]; }
      idxI[s + 1] = bi;
    }
  }
}

// -------------------------------------------------------------- gather ----
__global__ __launch_bounds__(TPB) void fps_gather(const float* __restrict__ feat,
                                                  const int* __restrict__ idxI,
                                                  float* __restrict__ out) {
  const int j = blockIdx.x;
  const int c = threadIdx.x;
  const int p = idxI[j];
  out[(size_t)j * C_DIM + c] = feat[(size_t)p * C_DIM + c];
  if (c == 0) out[(size_t)S_SAMP * C_DIM + j] = (float)p;
}

// ---------------------------------------------------------------- host ----
extern "C" void kernel_launch(void* const* d_in, const int* in_sizes, int n_in,
                              void* d_out, int out_size, void* d_ws, size_t ws_size,
                              hipStream_t stream) {
  const float* feat = (const float*)d_in[0];
  float* out = (float*)d_out;

  char* ws = (char*)d_ws;
  float* normx = (float*)(ws);                     // 400000 B
  float* mind2 = (float*)(ws + 400000);            // 400000 B
  int*   idxI  = (int*)  (ws + 800000);            // 4000 B
  int*   cnt   = (int*)  (ws + 804032);            // 4000 B
  float* bval  = (float*)(ws + 808064);            // NBLK*4
  int*   bidx  = (int*)  (ws + 811264);            // NBLK*4

  fps_init<<<(N_PTS + TPB - 1) / TPB, TPB, 0, stream>>>(feat, normx, mind2, idxI, cnt);

  for (int s = 0; s < S_SAMP - 1; ++s) {
    fps_step<<<NBLK, TPB, 0, stream>>>(feat, normx, mind2, idxI, cnt, bval, bidx, s);
  }

  fps_gather<<<S_SAMP, TPB, 0, stream>>>(feat, idxI, out);
}